// SAGE_17428977287481
// MI455X (gfx1250) — compile-verified
//
#include <hip/hip_runtime.h>

typedef float v2f __attribute__((ext_vector_type(2)));
typedef float v8f __attribute__((ext_vector_type(8)));

#define D 64

// ---------------------------------------------------------------------------
// Degree: deg[dst[e]] += 1  (graph is static; computed once, reused 3 layers)
// ---------------------------------------------------------------------------
__global__ void sage_deg_kernel(const int* __restrict__ dst,
                                float* __restrict__ deg, int nEdges) {
  int e = blockIdx.x * blockDim.x + threadIdx.x;
  if (e < nEdges) atomicAdd(deg + dst[e], 1.0f);
}

// ---------------------------------------------------------------------------
// Scatter-add: agg[dst[e]][:] += in[src[e]][:]
// 16 threads per edge, each moves 4 channels (float4 load, 4 f32 atomics).
// agg/x are L2-resident (25.6 MB each vs 192 MB L2) -> atomics resolve in L2.
// ---------------------------------------------------------------------------
__global__ void sage_scatter_kernel(const float* __restrict__ xin,
                                    const int* __restrict__ src,
                                    const int* __restrict__ dst,
                                    float* __restrict__ agg, int nEdges) {
  int idx = blockIdx.x * blockDim.x + threadIdx.x;
  int total = nEdges * 16;
  if (idx >= total) return;
  int e = idx >> 4;
  int c = (idx & 15) << 2;
  int s = src[e];
  int d = dst[e];
  const float4 v = *(const float4*)(xin + (size_t)s * D + c);
  float* p = agg + (size_t)d * D + c;
  atomicAdd(p + 0, v.x);
  atomicAdd(p + 1, v.y);
  atomicAdd(p + 2, v.z);
  atomicAdd(p + 3, v.w);
}

// ---------------------------------------------------------------------------
// Fused SAGE linear layer via fp32 WMMA (V_WMMA_F32_16X16X4_F32):
//   out[16 rows][64 cols] = relu?( (agg/deg) @ Wl^T + bl + x @ Wr^T )
// One wave per 16-row tile; 4 column tiles of 16; K=64 in 16 steps of 4.
// A (16x4 f32) per ISA: lanes 0-15 -> K={k,k+1}, lanes 16-31 -> K={k+2,k+3},
// row M = lane&15.  B (4x16): N = lane&15, same K split per half-wave.
// C/D: VGPR v -> M = v + 8*(lane>=16), N = lane&15.
// ---------------------------------------------------------------------------
template <bool RELU>
__global__ void sage_wmma_kernel(const float* __restrict__ agg,
                                 const float* __restrict__ deg,
                                 const float* __restrict__ xin,
                                 const float* __restrict__ Wl,
                                 const float* __restrict__ bl,
                                 const float* __restrict__ Wr,
                                 float* __restrict__ out, int nTiles) {
  const int lane = threadIdx.x & 31;
  const int wave = threadIdx.x >> 5;
  const int tile = blockIdx.x * (blockDim.x >> 5) + wave;
  if (tile >= nTiles) return;  // wave-uniform: EXEC stays all-ones for WMMA

  const int m = lane & 15;             // row within tile (A) / out column (B,D)
  const int khalf = (lane >> 4) << 1;  // 0 for lanes 0-15, 2 for lanes 16-31
  const int row0 = tile << 4;
  const int row = row0 + m;

  const float rdeg = 1.0f / fmaxf(deg[row], 1.0f);
  const float* __restrict__ arow = agg + (size_t)row * D;
  const float* __restrict__ xrow = xin + (size_t)row * D;

  v8f acc[4];
#pragma unroll
  for (int nt = 0; nt < 4; ++nt) acc[nt] = (v8f){0, 0, 0, 0, 0, 0, 0, 0};

#pragma unroll
  for (int kk = 0; kk < 16; ++kk) {
    const int k = (kk << 2) + khalf;
    v2f am = *(const v2f*)(arow + k);  // this lane's K-pair of mean row
    am[0] *= rdeg;
    am[1] *= rdeg;
    const v2f ax = *(const v2f*)(xrow + k);  // root term row
#pragma unroll
    for (int nt = 0; nt < 4; ++nt) {
      const int j = (nt << 4) + m;  // output column
      // B[k][j] = W[j][k]  (out = in @ W^T), contiguous in k -> v2f load
      const v2f bL = *(const v2f*)(Wl + j * D + k);
      const v2f bR = *(const v2f*)(Wr + j * D + k);
      acc[nt] = __builtin_amdgcn_wmma_f32_16x16x4_f32(
          false, am, false, bL, (short)0, acc[nt], false, false);
      acc[nt] = __builtin_amdgcn_wmma_f32_16x16x4_f32(
          false, ax, false, bR, (short)0, acc[nt], false, false);
    }
  }

  const int mhi = (lane >> 4) << 3;  // 0 or 8: D-matrix M offset per half-wave
#pragma unroll
  for (int nt = 0; nt < 4; ++nt) {
    const int j = (nt << 4) + m;
    const float b = bl[j];
#pragma unroll
    for (int v = 0; v < 8; ++v) {
      float val = acc[nt][v] + b;
      if (RELU) val = fmaxf(val, 0.0f);
      out[(size_t)(row0 + mhi + v) * D + j] = val;
    }
  }
}

// Scalar fallback for a row tail (unused for N=100000, kept for generality).
template <bool RELU>
__global__ void sage_tail_kernel(const float* __restrict__ agg,
                                 const float* __restrict__ deg,
                                 const float* __restrict__ xin,
                                 const float* __restrict__ Wl,
                                 const float* __restrict__ bl,
                                 const float* __restrict__ Wr,
                                 float* __restrict__ out, int row0, int nRows) {
  int idx = blockIdx.x * blockDim.x + threadIdx.x;
  if (idx >= nRows * D) return;
  int r = row0 + idx / D;
  int j = idx % D;
  float rdeg = 1.0f / fmaxf(deg[r], 1.0f);
  float s = bl[j];
  for (int k = 0; k < D; ++k)
    s += agg[(size_t)r * D + k] * rdeg * Wl[j * D + k] +
         xin[(size_t)r * D + k] * Wr[j * D + k];
  if (RELU) s = fmaxf(s, 0.0f);
  out[(size_t)r * D + j] = s;
}

static inline size_t align256(size_t x) { return (x + 255) & ~(size_t)255; }

extern "C" void kernel_launch(void* const* d_in, const int* in_sizes, int n_in,
                              void* d_out, int out_size, void* d_ws,
                              size_t ws_size, hipStream_t stream) {
  const float* x = (const float*)d_in[0];
  const int* src = (const int*)d_in[1];
  const int* dst = (const int*)d_in[2];
  const float* Wl[3] = {(const float*)d_in[3], (const float*)d_in[6],
                        (const float*)d_in[9]};
  const float* bl[3] = {(const float*)d_in[4], (const float*)d_in[7],
                        (const float*)d_in[10]};
  const float* Wr[3] = {(const float*)d_in[5], (const float*)d_in[8],
                        (const float*)d_in[11]};

  const int nNodes = in_sizes[0] / D;
  const int nEdges = in_sizes[1];
  const size_t featBytes = (size_t)nNodes * D * sizeof(float);

  // Workspace layout: deg | agg | h0 | h1   (~77 MB total)
  char* ws = (char*)d_ws;
  float* deg = (float*)ws;
  size_t ofs = align256((size_t)nNodes * sizeof(float));
  float* agg = (float*)(ws + ofs);
  ofs += align256(featBytes);
  float* h0 = (float*)(ws + ofs);
  ofs += align256(featBytes);
  float* h1 = (float*)(ws + ofs);

  // Degree once (graph static across layers)
  hipMemsetAsync(deg, 0, (size_t)nNodes * sizeof(float), stream);
  sage_deg_kernel<<<(nEdges + 255) / 256, 256, 0, stream>>>(dst, deg, nEdges);

  const int scatterThreads = nEdges * 16;
  const int scatterBlocks = (scatterThreads + 255) / 256;
  const int nTiles = nNodes / 16;
  const int gemmBlocks = (nTiles + 7) / 8;  // 8 waves (256 thr) per block
  const int tail = nNodes - nTiles * 16;

  auto run_layer = [&](const float* in, const float* wl, const float* b,
                       const float* wr, float* outp, bool relu) {
    hipMemsetAsync(agg, 0, featBytes, stream);
    sage_scatter_kernel<<<scatterBlocks, 256, 0, stream>>>(in, src, dst, agg,
                                                           nEdges);
    if (nTiles > 0) {
      if (relu)
        sage_wmma_kernel<true><<<gemmBlocks, 256, 0, stream>>>(
            agg, deg, in, wl, b, wr, outp, nTiles);
      else
        sage_wmma_kernel<false><<<gemmBlocks, 256, 0, stream>>>(
            agg, deg, in, wl, b, wr, outp, nTiles);
    }
    if (tail > 0) {
      int tb = (tail * D + 255) / 256;
      if (relu)
        sage_tail_kernel<true><<<tb, 256, 0, stream>>>(
            agg, deg, in, wl, b, wr, outp, nTiles * 16, tail);
      else
        sage_tail_kernel<false><<<tb, 256, 0, stream>>>(
            agg, deg, in, wl, b, wr, outp, nTiles * 16, tail);
    }
  };

  run_layer(x, Wl[0], bl[0], Wr[0], h0, /*relu=*/true);
  run_layer(h0, Wl[1], bl[1], Wr[1], h1, /*relu=*/true);
  run_layer(h1, Wl[2], bl[2], Wr[2], (float*)d_out, /*relu=*/false);
}